// GenericGraphEncoder_13469017441122
// MI455X (gfx1250) — compile-verified
//
#include <hip/hip_runtime.h>

// ---------------------------------------------------------------------------
// RGAT encoder for MI455X (gfx1250, wave32).
//  - Dense per-relation transforms use V_WMMA_F32_16X16X4_F32 (native fp32).
//  - Edge softmax/aggregation uses fp32 atomics resolved in the 192MB L2
//    (xr gather table = 38.4MB -> L2 resident).
// ---------------------------------------------------------------------------

typedef float v2f __attribute__((ext_vector_type(2)));
typedef float v8f __attribute__((ext_vector_type(8)));

#define HID 64
#define NREL 3
#define NLAYER 13
#define NEG_SLOPE 0.2f

// float atomic-max via signed/unsigned integer ordering trick
__device__ __forceinline__ void atomicMaxF(float* addr, float val) {
  if (val >= 0.0f) atomicMax((int*)addr, __float_as_int(val));
  else             atomicMin((unsigned int*)addr, __float_as_uint(val));
}

// ---------------------------------------------------------------------------
// xr[r][n][h] = sum_k x[n][k] * W[r][k][h]   via WMMA f32 16x16x4.
// One wave computes a 16(row) x 64(col) tile for one relation.
// A layout (16x4 f32): lanes 0-15 hold M=lane, VGPR0={K=0}, VGPR1={K=1};
//                      lanes 16-31 hold M=lane-16, VGPR0={K=2}, VGPR1={K=3}.
// B layout (4x16 f32): lanes 0-15 = N=lane with K=0/1 in v0/v1;
//                      lanes 16-31 = N=lane-16 with K=2/3.
// C/D layout: VGPR j, lanes 0-15 -> (M=j, N=lane); lanes 16-31 -> (M=j+8).
// ---------------------------------------------------------------------------
__global__ __launch_bounds__(256)
void k_rgemm(const float* __restrict__ x, int ldx, int D,
             const float* __restrict__ W,      // [NREL][D][HID]
             float* __restrict__ xr, int N)
{
  const int r    = blockIdx.y;
  const int wave = threadIdx.x >> 5;
  const int lane = threadIdx.x & 31;
  const int ntiles = (N + 15) >> 4;
  const int tile = blockIdx.x * 8 + wave;
  if (tile >= ntiles) return;                 // wave-uniform: EXEC stays all-1s
  const int row0 = tile << 4;
  const int half = lane >> 4;                 // 0: K=0/1 and M=j, 1: K=2/3 and M=j+8
  const int l16  = lane & 15;
  const int arow = row0 + l16;
  const float* Wr = W + (size_t)r * D * HID;

  v8f acc0 = {}, acc1 = {}, acc2 = {}, acc3 = {};
  for (int k = 0; k < D; k += 4) {
    const int ka = k + half * 2;
    v2f a;
    a.x = x[(size_t)arow * ldx + ka];
    a.y = x[(size_t)arow * ldx + ka + 1];
    const float* Wk0 = Wr + (size_t)ka * HID;
    const float* Wk1 = Wk0 + HID;
    v2f b0, b1, b2, b3;
    b0.x = Wk0[l16 +  0]; b0.y = Wk1[l16 +  0];
    b1.x = Wk0[l16 + 16]; b1.y = Wk1[l16 + 16];
    b2.x = Wk0[l16 + 32]; b2.y = Wk1[l16 + 32];
    b3.x = Wk0[l16 + 48]; b3.y = Wk1[l16 + 48];
    acc0 = __builtin_amdgcn_wmma_f32_16x16x4_f32(false, a, false, b0, (short)0, acc0, false, false);
    acc1 = __builtin_amdgcn_wmma_f32_16x16x4_f32(false, a, false, b1, (short)0, acc1, false, false);
    acc2 = __builtin_amdgcn_wmma_f32_16x16x4_f32(false, a, false, b2, (short)0, acc2, false, false);
    acc3 = __builtin_amdgcn_wmma_f32_16x16x4_f32(false, a, false, b3, (short)0, acc3, false, false);
  }

  float* out = xr + ((size_t)r * N + row0) * HID;
#pragma unroll
  for (int j = 0; j < 8; ++j) {
    const int rr = j + half * 8;
    out[(size_t)rr * HID + l16 +  0] = acc0[j];
    out[(size_t)rr * HID + l16 + 16] = acc1[j];
    out[(size_t)rr * HID + l16 + 32] = acc2[j];
    out[(size_t)rr * HID + l16 + 48] = acc3[j];
  }
}

// qn[i] = xr_row[i] . q ; kn[i] = xr_row[i] . k     (i over NREL*N rows)
__global__ void k_qk(const float* __restrict__ xr, const float* __restrict__ q,
                     const float* __restrict__ kv, float* __restrict__ qn,
                     float* __restrict__ kn, int total)
{
  int i = blockIdx.x * blockDim.x + threadIdx.x;
  if (i >= total) return;
  const float4* xp = (const float4*)(xr + (size_t)i * HID);
  const float4* qp = (const float4*)q;
  const float4* kp = (const float4*)kv;
  float sq = 0.f, sk = 0.f;
#pragma unroll
  for (int j = 0; j < HID / 4; ++j) {
    float4 v = xp[j], a = qp[j], b = kp[j];
    sq += v.x*a.x + v.y*a.y + v.z*a.z + v.w*a.w;
    sk += v.x*b.x + v.y*b.y + v.z*b.z + v.w*b.w;
  }
  qn[i] = sq; kn[i] = sk;
}

// out[n][h] = bias[h]; maxb[n] = -inf; sumb[n] = 0
__global__ void k_init_node(float* __restrict__ out, int ldo,
                            const float* __restrict__ bias,
                            float* __restrict__ maxb, float* __restrict__ sumb, int N)
{
  int i = blockIdx.x * blockDim.x + threadIdx.x;
  if (i >= N * HID) return;
  int n = i >> 6, h = i & 63;
  out[(size_t)n * ldo + h] = bias[h];
  if (h == 0) { maxb[n] = -__builtin_inff(); sumb[n] = 0.f; }
}

__global__ void k_edge_logit(const int* __restrict__ src, const int* __restrict__ dst,
                             const int* __restrict__ et,
                             const float* __restrict__ qn, const float* __restrict__ kn,
                             float* __restrict__ ebuf, float* __restrict__ maxb,
                             int E, int N)
{
  int e = blockIdx.x * blockDim.x + threadIdx.x;
  if (e >= E) return;
  int r = et[e], d = dst[e];
  float l = qn[(size_t)r * N + d] + kn[(size_t)r * N + src[e]];
  l = (l > 0.f) ? l : NEG_SLOPE * l;
  ebuf[e] = l;
  atomicMaxF(&maxb[d], l);
}

__global__ void k_edge_exp(const int* __restrict__ dst, float* __restrict__ ebuf,
                           const float* __restrict__ maxb, float* __restrict__ sumb, int E)
{
  int e = blockIdx.x * blockDim.x + threadIdx.x;
  if (e >= E) return;
  int d = dst[e];
  float v = expf(ebuf[e] - maxb[d]);
  ebuf[e] = v;                       // ebuf now holds exp values
  atomicAdd(&sumb[d], v);
}

// 16 threads per edge, one float4 of the 64-wide message each.
__global__ void k_edge_scatter(const int* __restrict__ src, const int* __restrict__ dst,
                               const int* __restrict__ et, const float* __restrict__ ebuf,
                               const float* __restrict__ sumb, const float* __restrict__ xr,
                               float* __restrict__ out, int ldo, int E, int N)
{
  long long idx = (long long)blockIdx.x * blockDim.x + threadIdx.x;
  int e = (int)(idx >> 4);
  if (e >= E) return;
  int q4 = (int)(idx & 15);
  int d = dst[e];
  float alpha = ebuf[e] / (sumb[d] + 1e-16f);
  const float4 v = *(const float4*)(xr + ((size_t)et[e] * N + src[e]) * HID + q4 * 4);
  float* op = out + (size_t)d * ldo + q4 * 4;
  atomicAdd(op + 0, alpha * v.x);
  atomicAdd(op + 1, alpha * v.y);
  atomicAdd(op + 2, alpha * v.z);
  atomicAdd(op + 3, alpha * v.w);
}

// ----- graph-level softmax aggregation ------------------------------------
__global__ void k_ginit(float* __restrict__ gmax, float* __restrict__ gsum,
                        float* __restrict__ grep, int total)
{
  int i = blockIdx.x * blockDim.x + threadIdx.x;
  if (i >= total) return;
  gmax[i] = -__builtin_inff(); gsum[i] = 0.f; grep[i] = 0.f;
}

__global__ void k_gmax(const float* __restrict__ rep, const int* __restrict__ bidx,
                       const float* __restrict__ t, float* __restrict__ gmax,
                       int N, int F)
{
  long long i = (long long)blockIdx.x * blockDim.x + threadIdx.x;
  if (i >= (long long)N * F) return;
  int n = (int)(i / F), f = (int)(i % F);
  atomicMaxF(&gmax[(size_t)bidx[n] * F + f], rep[i] * t[0]);
}

__global__ void k_gsum(const float* __restrict__ rep, const int* __restrict__ bidx,
                       const float* __restrict__ t, const float* __restrict__ gmax,
                       float* __restrict__ gsum, int N, int F)
{
  long long i = (long long)blockIdx.x * blockDim.x + threadIdx.x;
  if (i >= (long long)N * F) return;
  int n = (int)(i / F), f = (int)(i % F);
  size_t gi = (size_t)bidx[n] * F + f;
  float m = gmax[gi];
  if (m == -__builtin_inff()) m = 0.f;     // empty-segment neutralization
  atomicAdd(&gsum[gi], expf(rep[i] * t[0] - m));
}

__global__ void k_gout(const float* __restrict__ rep, const int* __restrict__ bidx,
                       const float* __restrict__ t, const float* __restrict__ gmax,
                       const float* __restrict__ gsum, float* __restrict__ grep,
                       int N, int F)
{
  long long i = (long long)blockIdx.x * blockDim.x + threadIdx.x;
  if (i >= (long long)N * F) return;
  int n = (int)(i / F), f = (int)(i % F);
  size_t gi = (size_t)bidx[n] * F + f;
  float m = gmax[gi];
  if (m == -__builtin_inff()) m = 0.f;
  float r = rep[i];
  float alpha = expf(r * t[0] - m) / (gsum[gi] + 1e-16f);
  atomicAdd(&grep[gi], r * alpha);
}

// ---------------------------------------------------------------------------
extern "C" void kernel_launch(void* const* d_in, const int* in_sizes, int n_in,
                              void* d_out, int out_size, void* d_ws, size_t ws_size,
                              hipStream_t stream)
{
  const float* x0    = (const float*)d_in[0];
  const int*   eidx  = (const int*)  d_in[1];
  const int*   etype = (const int*)  d_in[2];
  const int*   bidx  = (const int*)  d_in[3];
  const float* Wf    = (const float*)d_in[4];
  const float* qf    = (const float*)d_in[5];
  const float* kf    = (const float*)d_in[6];
  const float* bf    = (const float*)d_in[7];
  const float* Ws    = (const float*)d_in[8];
  const float* qs    = (const float*)d_in[9];
  const float* ks    = (const float*)d_in[10];
  const float* bs    = (const float*)d_in[11];
  const float* tS    = (const float*)d_in[12];

  const int N      = in_sizes[3];
  const int E      = in_sizes[2];
  const int IN_DIM = in_sizes[0] / N;
  const int F      = NLAYER * HID;            // 832
  const int G      = out_size / F - N;

  const int* src = eidx;
  const int* dst = eidx + E;

  // workspace carve-up (floats, 256B-aligned chunks)
  float* ws = (float*)d_ws;
  size_t off = 0;
  auto carve = [&](size_t cnt) { float* p = ws + off; off += (cnt + 63) & ~(size_t)63; return p; };
  float* xr   = carve((size_t)NREL * N * HID);   // 38.4 MB: L2-resident gather table
  float* qn   = carve((size_t)NREL * N);
  float* kn   = carve((size_t)NREL * N);
  float* maxb = carve((size_t)N);
  float* sumb = carve((size_t)N);
  float* ebuf = carve((size_t)E);
  float* gmax = carve((size_t)G * F);
  float* gsum = carve((size_t)G * F);

  float* grep = (float*)d_out;                   // [G, 832]
  float* nrep = (float*)d_out + (size_t)G * F;   // [N, 832], layer l at col l*64

  const int ntilesM = (N + 15) / 16;
  dim3 ggrid((ntilesM + 7) / 8, NREL);

  for (int l = 0; l < NLAYER; ++l) {
    const float *x, *W, *q, *k, *b; int ldx, D;
    if (l == 0) { x = x0; ldx = IN_DIM; D = IN_DIM; W = Wf; q = qf; k = kf; b = bf; }
    else {
      x  = nrep + (size_t)(l - 1) * HID; ldx = F; D = HID;
      W  = Ws + (size_t)(l - 1) * NREL * HID * HID;
      q  = qs + (size_t)(l - 1) * HID;
      k  = ks + (size_t)(l - 1) * HID;
      b  = bs + (size_t)(l - 1) * HID;
    }
    float* out = nrep + (size_t)l * HID;

    k_rgemm<<<ggrid, 256, 0, stream>>>(x, ldx, D, W, xr, N);
    const int tq = NREL * N;
    k_qk<<<(tq + 255) / 256, 256, 0, stream>>>(xr, q, k, qn, kn, tq);
    k_init_node<<<(N * HID + 255) / 256, 256, 0, stream>>>(out, F, b, maxb, sumb, N);
    k_edge_logit<<<(E + 255) / 256, 256, 0, stream>>>(src, dst, etype, qn, kn, ebuf, maxb, E, N);
    k_edge_exp<<<(E + 255) / 256, 256, 0, stream>>>(dst, ebuf, maxb, sumb, E);
    const long long st = (long long)E * 16;
    k_edge_scatter<<<(unsigned)((st + 255) / 256), 256, 0, stream>>>(src, dst, etype, ebuf,
                                                                     sumb, xr, out, F, E, N);
  }

  const int gt = G * F;
  k_ginit<<<(gt + 255) / 256, 256, 0, stream>>>(gmax, gsum, grep, gt);
  const long long nf = (long long)N * F;
  const unsigned nb = (unsigned)((nf + 255) / 256);
  k_gmax<<<nb, 256, 0, stream>>>(nrep, bidx, tS, gmax, N, F);
  k_gsum<<<nb, 256, 0, stream>>>(nrep, bidx, tS, gmax, gsum, N, F);
  k_gout<<<nb, 256, 0, stream>>>(nrep, bidx, tS, gmax, gsum, grep, N, F);
}